// EdgeConvModule_10316511445758
// MI455X (gfx1250) — compile-verified
//
#include <hip/hip_runtime.h>
#include <hip/hip_bf16.h>

typedef __attribute__((ext_vector_type(2))) float v2f;
typedef __attribute__((ext_vector_type(4))) float v4f;
typedef __attribute__((ext_vector_type(8))) float v8f;
typedef __attribute__((ext_vector_type(4))) int   v4i;

#define NB   4
#define NCH  64
#define NPTS 8192
#define KNN  20
#define NEG_INF (-3.4e38f)

#define AS1 __attribute__((address_space(1)))
#define AS3 __attribute__((address_space(3)))

// ---- CDNA5 async global->LDS (ASYNCcnt) with synchronous fallback ----------
#if defined(__has_builtin)
#if __has_builtin(__builtin_amdgcn_global_load_async_to_lds_b128) && \
    __has_builtin(__builtin_amdgcn_global_load_async_to_lds_b32)  && \
    __has_builtin(__builtin_amdgcn_s_wait_asynccnt)
#define HAVE_ASYNC 1
#endif
#endif
#ifndef HAVE_ASYNC
#define HAVE_ASYNC 0
#endif

__device__ __forceinline__ void cp16(float* l, const float* g) {
#if HAVE_ASYNC
    __builtin_amdgcn_global_load_async_to_lds_b128(
        (AS1 v4i*)(AS1 void*)g, (AS3 v4i*)(AS3 void*)l, 0, 0);
#else
    *(v4f*)l = *(const v4f*)g;
#endif
}
__device__ __forceinline__ void cp4(float* l, const float* g) {
#if HAVE_ASYNC
    __builtin_amdgcn_global_load_async_to_lds_b32(
        (AS1 int*)(AS1 void*)g, (AS3 int*)(AS3 void*)l, 0, 0);
#else
    *l = *g;
#endif
}
__device__ __forceinline__ void wait_async() {
#if HAVE_ASYNC
    __builtin_amdgcn_s_wait_asynccnt(0);
#endif
}

// ---------------------------------------------------------------------------
// Kernel 0: per-point squared norms. input layout (B, C, N).
// ---------------------------------------------------------------------------
__global__ __launch_bounds__(256) void sqnorm_kernel(const float* __restrict__ x,
                                                     float* __restrict__ sq) {
    int gid = blockIdx.x * 256 + threadIdx.x;
    int b = gid / NPTS, n = gid % NPTS;
    const float* xb = x + (size_t)b * NCH * NPTS;
    float s = 0.f;
#pragma unroll 8
    for (int c = 0; c < NCH; ++c) {
        float v = xb[c * NPTS + n];
        s = fmaf(v, v, s);
    }
    sq[gid] = s;
}

// ---------------------------------------------------------------------------
// Kernel 1: KNN. WG = 256 thr (8 waves) owns 128 rows. Channel-major LDS
// tiles; A fragments hoisted to registers; 64-wide column tiles double-
// buffered via async-to-LDS; dual accumulators for WMMA-chain ILP.
// ---------------------------------------------------------------------------
#define XCS 72          // Xc LDS row stride (8k+col banks -> halves disjoint)
#define SST 66          // score tile stride

__global__ __launch_bounds__(256) void knn_kernel(const float* __restrict__ x,
                                                  const float* __restrict__ sq,
                                                  int* __restrict__ knn_out) {
    const int b    = blockIdx.y;
    const int row0 = blockIdx.x * 128;
    extern __shared__ float smem[];
    float* XrL = smem;                    // [64][128]  chan-major row tile
    float* XcL = XrL + 64 * 128;          // 2 x [64][72] column tiles
    float* S   = XcL + 2 * 64 * XCS;      // [128][66]
    float* sqc = S + 128 * SST;           // 2 x [64]

    const int t    = threadIdx.x;
    const int lane = t & 31;
    const int wave = t >> 5;
    const int n0   = lane & 15;
    const int half = lane >> 4;
    const int m_base = wave * 16;
    const int r = t & 127;                // scan row
    const int h = t >> 7;                 // scan column half (32 cols)
    const float* xb  = x + (size_t)b * NCH * NPTS;
    const float* sqb = sq + (size_t)b * NPTS;

    // Row tile: straight copy (point index contiguous per channel row).
#pragma unroll
    for (int i = 0; i < 8; ++i) {
        int e = t * 4 + i * 1024, c = e >> 7, p = e & 127;
        cp16(&XrL[c * 128 + p], &xb[c * NPTS + row0 + p]);
    }
    // Column tile 0 into buffer 0.
#pragma unroll
    for (int i = 0; i < 4; ++i) {
        int e = t * 4 + i * 1024, c = e >> 6, p = e & 63;
        cp16(&XcL[c * XCS + p], &xb[c * NPTS + p]);
    }
    if (t < 64) cp4(&sqc[t], &sqb[t]);
    wait_async();
    __syncthreads();

    // Hoist A fragments: wave's 16-row strip, all 16 K-steps. Loaded once.
    v2f af[16];
#pragma unroll
    for (int k = 0; k < 16; ++k) {
        int c0 = 4 * k + 2 * half;
        af[k].x = XrL[c0 * 128 + m_base + n0];
        af[k].y = XrL[(c0 + 1) * 128 + m_base + n0];
    }

    float bv[KNN]; int bi[KNN];
#pragma unroll
    for (int i = 0; i < KNN; ++i) { bv[i] = NEG_INF; bi[i] = 0; }
    float minv = NEG_INF;

    for (int ct = 0; ct < NPTS / 64; ++ct) {
        const int cur = ct & 1, nxt = cur ^ 1;
        const int col0 = ct * 64;

        // Prefetch next column tile into the other buffer (async).
        if (ct + 1 < NPTS / 64) {
            const int nc0 = col0 + 64;
#pragma unroll
            for (int i = 0; i < 4; ++i) {
                int e = t * 4 + i * 1024, c = e >> 6, p = e & 63;
                cp16(&XcL[nxt * 64 * XCS + c * XCS + p], &xb[c * NPTS + nc0 + p]);
            }
            if (t < 64) cp4(&sqc[nxt * 64 + t], &sqb[nc0 + t]);
        }

        // WMMA on current buffer: 16-row strip x 4 column blocks, processed
        // in pairs with dual accumulators for dependency-chain ILP.
        const float* Xcb = &XcL[cur * 64 * XCS];
        for (int cbp = 0; cbp < 2; ++cbp) {
            const int cc0 = (2 * cbp + 0) * 16 + n0;
            const int cc1 = (2 * cbp + 1) * 16 + n0;
            v8f acc0 = {}, acc1 = {};
#pragma unroll
            for (int k = 0; k < 16; ++k) {
                int c0 = 4 * k + 2 * half;
                v2f b0, b1;
                b0.x = Xcb[c0 * XCS + cc0];
                b0.y = Xcb[(c0 + 1) * XCS + cc0];
                b1.x = Xcb[c0 * XCS + cc1];
                b1.y = Xcb[(c0 + 1) * XCS + cc1];
                acc0 = __builtin_amdgcn_wmma_f32_16x16x4_f32(
                    false, af[k], false, b0, (short)0, acc0, false, false);
                acc1 = __builtin_amdgcn_wmma_f32_16x16x4_f32(
                    false, af[k], false, b1, (short)0, acc1, false, false);
            }
#pragma unroll
            for (int rr = 0; rr < 8; ++rr) {
                S[(m_base + rr + 8 * half) * SST + (2 * cbp + 0) * 16 + n0] = acc0[rr];
                S[(m_base + rr + 8 * half) * SST + (2 * cbp + 1) * 16 + n0] = acc1[rr];
            }
        }
        __syncthreads();   // S complete

        // Scan: thread owns (row r, 32 columns).
        const float* Srow = &S[r * SST + h * 32];
        const float* sqh  = &sqc[cur * 64 + h * 32];
        const int jbase   = col0 + h * 32;
        for (int c = 0; c < 32; ++c) {
            float sval = fmaf(2.0f, Srow[c], -sqh[c]);
            if (sval > minv) {
                int mp = 0; float mv = bv[0];
#pragma unroll
                for (int i = 1; i < KNN; ++i)
                    if (bv[i] < mv) { mv = bv[i]; mp = i; }
#pragma unroll
                for (int i = 0; i < KNN; ++i)
                    if (i == mp) { bv[i] = sval; bi[i] = jbase + c; }
                float nm = bv[0];
#pragma unroll
                for (int i = 1; i < KNN; ++i) nm = fminf(nm, bv[i]);
                minv = nm;
            }
        }
        wait_async();      // next buffer landed (this wave's issues)
        __syncthreads();   // all waves: scan done, next tile valid
    }

    // Merge per-row half-lists (reuse S / Xc regions).
    float* Lval = S;                 // [128][2][KNN] floats
    int*   Lidx = (int*)XcL;         // [128][2][KNN] ints
#pragma unroll
    for (int i = 0; i < KNN; ++i) {
        Lval[(r * 2 + h) * KNN + i] = bv[i];
        Lidx[(r * 2 + h) * KNN + i] = bi[i];
    }
    __syncthreads();
    if (t < 128) {
        unsigned long long taken = 0ull;
        int* outp = knn_out + ((size_t)b * NPTS + row0 + t) * KNN;
        const float* lv = &Lval[t * 2 * KNN];
        const int*   li = &Lidx[t * 2 * KNN];
        for (int k = 0; k < KNN; ++k) {
            float best = NEG_INF; int bc = 0;
            for (int cc = 0; cc < 2 * KNN; ++cc) {
                if (!((taken >> cc) & 1ull)) {
                    float v = lv[cc];
                    if (v > best) { best = v; bc = cc; }
                }
            }
            taken |= (1ull << bc);
            outp[k] = li[bc];
        }
    }
}

// ---------------------------------------------------------------------------
// Kernel 2: T[b][n][0:64] = Wl.x_n, T[b][n][64:128] = Wr.x_n (collapsed edge
// einsum). Same WMMA tiling, A hoisted, dual accumulators.
// ---------------------------------------------------------------------------
#define WCS 136

__global__ __launch_bounds__(256) void uv_gemm_kernel(const float* __restrict__ x,
                                                      const float* __restrict__ W,
                                                      float* __restrict__ T) {
    const int b    = blockIdx.y;
    const int row0 = blockIdx.x * 128;
    extern __shared__ float smem[];
    float* XrL = smem;             // [64][128] chan-major
    float* WcL = XrL + 64 * 128;   // [64][136] chan x out'

    const int t = threadIdx.x;
    const float* xb = x + (size_t)b * NCH * NPTS;
#pragma unroll
    for (int i = 0; i < 8; ++i) {
        int e = t * 4 + i * 1024, c = e >> 7, p = e & 127;
        cp16(&XrL[c * 128 + p], &xb[c * NPTS + row0 + p]);
    }
    for (int e = t; e < 64 * 128; e += 256) {
        int o = e & 127, c = e >> 7;
        WcL[c * WCS + o] = (o < 64) ? W[o * 128 + c] : W[(o - 64) * 128 + 64 + c];
    }
    wait_async();
    __syncthreads();

    const int lane = t & 31;
    const int wave = t >> 5;
    const int n0   = lane & 15;
    const int half = lane >> 4;
    const int m_base = wave * 16;

    v2f af[16];
#pragma unroll
    for (int k = 0; k < 16; ++k) {
        int c0 = 4 * k + 2 * half;
        af[k].x = XrL[c0 * 128 + m_base + n0];
        af[k].y = XrL[(c0 + 1) * 128 + m_base + n0];
    }

    for (int cbp = 0; cbp < 4; ++cbp) {
        const int cc0 = (2 * cbp + 0) * 16 + n0;
        const int cc1 = (2 * cbp + 1) * 16 + n0;
        v8f acc0 = {}, acc1 = {};
#pragma unroll
        for (int k = 0; k < 16; ++k) {
            int c0 = 4 * k + 2 * half;
            v2f b0, b1;
            b0.x = WcL[c0 * WCS + cc0];
            b0.y = WcL[(c0 + 1) * WCS + cc0];
            b1.x = WcL[c0 * WCS + cc1];
            b1.y = WcL[(c0 + 1) * WCS + cc1];
            acc0 = __builtin_amdgcn_wmma_f32_16x16x4_f32(
                false, af[k], false, b0, (short)0, acc0, false, false);
            acc1 = __builtin_amdgcn_wmma_f32_16x16x4_f32(
                false, af[k], false, b1, (short)0, acc1, false, false);
        }
#pragma unroll
        for (int rr = 0; rr < 8; ++rr) {
            int n = row0 + m_base + rr + 8 * half;
            float* Trow = T + ((size_t)b * NPTS + n) * 128;
            Trow[(2 * cbp + 0) * 16 + n0] = acc0[rr];
            Trow[(2 * cbp + 1) * 16 + n0] = acc1[rr];
        }
    }
}

// ---------------------------------------------------------------------------
// Kernel 3: out[b][o][n] = leaky((max_k u[j_k][o] - u[n][o] + v[n][o])*inv+sh)
// ---------------------------------------------------------------------------
__global__ __launch_bounds__(256) void edge_max_kernel(const float* __restrict__ T,
                                                       const int* __restrict__ knn_idx,
                                                       const float* __restrict__ gamma,
                                                       const float* __restrict__ beta,
                                                       const float* __restrict__ rmean,
                                                       const float* __restrict__ rvar,
                                                       float* __restrict__ out) {
    const int b  = blockIdx.y;
    const int p0 = blockIdx.x * 64;
    __shared__ float outs[64 * 65];
    __shared__ int   idxs[64 * KNN];

    const int t = threadIdx.x;
    for (int e = t; e < 64 * KNN; e += 256)
        idxs[e] = knn_idx[((size_t)b * NPTS + p0) * KNN + e];
    __syncthreads();

    const int p  = t >> 2;
    const int oq = (t & 3) * 16;
    const float* Tb    = T + (size_t)b * NPTS * 128;
    const float* Tself = Tb + (size_t)(p0 + p) * 128;

    float m[16];
#pragma unroll
    for (int i = 0; i < 16; ++i) m[i] = NEG_INF;
    for (int k = 0; k < KNN; ++k) {
        int j = idxs[p * KNN + k];
        const float* Tj = Tb + (size_t)j * 128 + oq;
#pragma unroll
        for (int q = 0; q < 4; ++q) {
            v4f tv = *(const v4f*)(Tj + 4 * q);
            m[4 * q + 0] = fmaxf(m[4 * q + 0], tv.x);
            m[4 * q + 1] = fmaxf(m[4 * q + 1], tv.y);
            m[4 * q + 2] = fmaxf(m[4 * q + 2], tv.z);
            m[4 * q + 3] = fmaxf(m[4 * q + 3], tv.w);
        }
    }
#pragma unroll
    for (int i = 0; i < 16; ++i) {
        int o = oq + i;
        float inv = gamma[o] * rsqrtf(rvar[o] + 1e-5f);
        float sh  = fmaf(-rmean[o], inv, beta[o]);
        float y   = m[i] - Tself[o] + Tself[64 + o];
        y = fmaf(y, inv, sh);
        y = (y >= 0.f) ? y : 0.2f * y;
        outs[o * 65 + p] = y;
    }
    __syncthreads();
    for (int e = t; e < 64 * 64; e += 256) {
        int o = e >> 6, pp = e & 63;
        out[((size_t)b * 64 + o) * NPTS + p0 + pp] = outs[o * 65 + pp];
    }
}

// ---------------------------------------------------------------------------
extern "C" void kernel_launch(void* const* d_in, const int* in_sizes, int n_in,
                              void* d_out, int out_size, void* d_ws, size_t ws_size,
                              hipStream_t stream) {
    const float* x     = (const float*)d_in[0];
    const float* W     = (const float*)d_in[1];
    const float* gamma = (const float*)d_in[2];
    const float* beta  = (const float*)d_in[3];
    const float* rmean = (const float*)d_in[4];
    const float* rvar  = (const float*)d_in[5];
    float* out = (float*)d_out;

    char* ws = (char*)d_ws;
    float* sq  = (float*)ws;                                   // B*N floats
    int*   knn = (int*)(ws + (size_t)NB * NPTS * 4);           // B*N*K ints
    float* T   = (float*)(ws + (size_t)NB * NPTS * 4
                             + (size_t)NB * NPTS * KNN * 4);   // B*N*128 floats

    sqnorm_kernel<<<NB * NPTS / 256, 256, 0, stream>>>(x, sq);

    size_t knn_smem = (size_t)(64 * 128 + 2 * 64 * XCS + 128 * SST + 128)
                      * sizeof(float);
    knn_kernel<<<dim3(NPTS / 128, NB), 256, knn_smem, stream>>>(x, sq, knn);

    size_t gemm_smem = (size_t)(64 * 128 + 64 * WCS) * sizeof(float);
    uv_gemm_kernel<<<dim3(NPTS / 128, NB), 256, gemm_smem, stream>>>(x, W, T);

    edge_max_kernel<<<dim3(NPTS / 64, NB), 256, 0, stream>>>(T, knn, gamma, beta,
                                                             rmean, rvar, out);
}